// NodeClassifier_49641232007443
// MI455X (gfx1250) — compile-verified
//
#include <hip/hip_runtime.h>
#include <hip/hip_bf16.h>
#include <math.h>

typedef __attribute__((ext_vector_type(2))) float v2f;
typedef __attribute__((ext_vector_type(8))) float v8f;

#define SELU_SCALE 1.0507009873554805f
#define SELU_ALPHA 1.6732632423543772f

__device__ __forceinline__ v8f wmma4(v2f a, v2f b, v8f c) {
    return __builtin_amdgcn_wmma_f32_16x16x4_f32(false, a, false, b, (short)0, c, false, false);
}

// ---------------- degree ----------------
__global__ void deg_kernel(const int* __restrict__ dst, float* __restrict__ deg, int E) {
    int e = blockIdx.x * blockDim.x + threadIdx.x;
    if (e < E) atomicAdd(&deg[dst[e]], 1.0f);
}

// ---------------- norm: dis = rsqrt(deg) (0 if deg==0), cntinv = 1/max(deg,1) ----------------
__global__ void norm_kernel(const float* __restrict__ deg, float* __restrict__ dis,
                            float* __restrict__ cntinv, int n) {
    int i = blockIdx.x * blockDim.x + threadIdx.x;
    if (i < n) {
        float d = deg[i];
        dis[i]    = (d > 0.0f) ? rsqrtf(fmaxf(d, 1.0f)) : 0.0f;
        cntinv[i] = 1.0f / fmaxf(d, 1.0f);
    }
}

// ---------------- edge propagation: Y[dst] += X[src] * w ----------------
// 16 lanes per edge, float4 per lane (64 feats). weighted: w = dis[dst]*dis[src], else 1.
__global__ void prop_kernel(const float* __restrict__ X, float* __restrict__ Y,
                            const int* __restrict__ src, const int* __restrict__ dst,
                            const float* __restrict__ dis, int E, int weighted) {
    int tid = blockIdx.x * blockDim.x + threadIdx.x;
    int e   = tid >> 4;
    int seg = tid & 15;
    if (e >= E) return;
    int s = src[e];
    int t = dst[e];
    float w = 1.0f;
    if (weighted) w = dis[s] * dis[t];
    const float4 v = ((const float4*)(X + (long)s * 64))[seg];
    float* yr = Y + (long)t * 64 + seg * 4;
    atomicAdd(yr + 0, v.x * w);
    atomicAdd(yr + 1, v.y * w);
    atomicAdd(yr + 2, v.z * w);
    atomicAdd(yr + 3, v.w * w);
}

// ---------------- GEMM1: Out = selu( (Sagg*cntinv) @ Wl (64x64) + H @ Wr (64x64) + bias ) ----------------
// One wave computes a full 16x64 row block: 4 independent accumulator chains,
// A fragments (agg/h rows) loaded once and reused across all 4 column tiles.
__global__ __launch_bounds__(256) void gemm1_kernel(
    const float* __restrict__ Sagg, const float* __restrict__ Hf,
    const float* __restrict__ cntinv,
    const float* __restrict__ Wl, const float* __restrict__ Wr,
    const float* __restrict__ bias, float* __restrict__ Out, int nnodes)
{
    int wave = (blockIdx.x * blockDim.x + threadIdx.x) >> 5;
    int lane = threadIdx.x & 31;
    int rb = wave;                              // 16-row block
    int rbCount = (nnodes + 15) >> 4;
    if (rb >= rbCount) return;                  // wave-uniform
    int m = rb * 16 + (lane & 15);
    int mclamp = m < nnodes ? m : (nnodes - 1); // safe loads for ragged tail
    int col0 = lane & 15;
    int kshift = (lane >> 4) * 2;               // 0 for lanes 0-15, 2 for 16-31
    float cinv = cntinv[mclamp];
    const float* aggRow = Sagg + (long)mclamp * 64;
    const float* hRow   = Hf   + (long)mclamp * 64;
    v8f acc0 = {}, acc1 = {}, acc2 = {}, acc3 = {};
    #pragma unroll 4
    for (int k0 = 0; k0 < 64; k0 += 4) {
        int kk = k0 + kshift;
        v2f a1; a1.x = aggRow[kk] * cinv; a1.y = aggRow[kk + 1] * cinv;
        v2f a2; a2.x = hRow[kk];          a2.y = hRow[kk + 1];
        const float* wl0 = Wl + kk * 64 + col0;
        const float* wl1 = Wl + (kk + 1) * 64 + col0;
        const float* wr0 = Wr + kk * 64 + col0;
        const float* wr1 = Wr + (kk + 1) * 64 + col0;
        v2f b;
        b.x = wl0[ 0]; b.y = wl1[ 0]; acc0 = wmma4(a1, b, acc0);
        b.x = wl0[16]; b.y = wl1[16]; acc1 = wmma4(a1, b, acc1);
        b.x = wl0[32]; b.y = wl1[32]; acc2 = wmma4(a1, b, acc2);
        b.x = wl0[48]; b.y = wl1[48]; acc3 = wmma4(a1, b, acc3);
        b.x = wr0[ 0]; b.y = wr1[ 0]; acc0 = wmma4(a2, b, acc0);
        b.x = wr0[16]; b.y = wr1[16]; acc1 = wmma4(a2, b, acc1);
        b.x = wr0[32]; b.y = wr1[32]; acc2 = wmma4(a2, b, acc2);
        b.x = wr0[48]; b.y = wr1[48]; acc3 = wmma4(a2, b, acc3);
    }
    int rowbase = rb * 16 + (lane >> 4) * 8;
    v8f accs[4] = {acc0, acc1, acc2, acc3};
    #pragma unroll
    for (int ct = 0; ct < 4; ++ct) {
        int col = ct * 16 + col0;
        float bn = bias[col];
        #pragma unroll
        for (int j = 0; j < 8; ++j) {
            int row = rowbase + j;
            float v = accs[ct][j] + bn;
            v = v > 0.0f ? SELU_SCALE * v : SELU_SCALE * SELU_ALPHA * (expf(v) - 1.0f);
            if (row < nnodes) Out[(long)row * 64 + col] = v;
        }
    }
}

// ---------------- GEMM2: Logits = (Sagg*cntinv) @ Wl (64x40) + H @ Wr (64x40) + bias ----------------
// One wave computes 16x48 (40 valid cols padded to 3 tiles); 3 independent chains.
__global__ __launch_bounds__(256) void gemm2_kernel(
    const float* __restrict__ Sagg, const float* __restrict__ Hf,
    const float* __restrict__ cntinv,
    const float* __restrict__ Wl, const float* __restrict__ Wr,
    const float* __restrict__ bias, float* __restrict__ Logits, int nnodes)
{
    int wave = (blockIdx.x * blockDim.x + threadIdx.x) >> 5;
    int lane = threadIdx.x & 31;
    int rb = wave;
    int rbCount = (nnodes + 15) >> 4;
    if (rb >= rbCount) return;                  // wave-uniform
    int m = rb * 16 + (lane & 15);
    int mclamp = m < nnodes ? m : (nnodes - 1);
    int col0 = lane & 15;
    int kshift = (lane >> 4) * 2;
    float cinv = cntinv[mclamp];
    // tile 2 covers cols 32..47; only col < 40 valid -> branchless zero-mask
    int col2  = 32 + col0;
    int c2off = col2 < 40 ? col2 : 0;
    float msk2 = col2 < 40 ? 1.0f : 0.0f;
    const float* aggRow = Sagg + (long)mclamp * 64;
    const float* hRow   = Hf   + (long)mclamp * 64;
    v8f acc0 = {}, acc1 = {}, acc2 = {};
    #pragma unroll 4
    for (int k0 = 0; k0 < 64; k0 += 4) {
        int kk = k0 + kshift;
        v2f a1; a1.x = aggRow[kk] * cinv; a1.y = aggRow[kk + 1] * cinv;
        v2f a2; a2.x = hRow[kk];          a2.y = hRow[kk + 1];
        const float* wl0 = Wl + kk * 40;
        const float* wl1 = Wl + (kk + 1) * 40;
        const float* wr0 = Wr + kk * 40;
        const float* wr1 = Wr + (kk + 1) * 40;
        v2f b;
        b.x = wl0[col0];          b.y = wl1[col0];          acc0 = wmma4(a1, b, acc0);
        b.x = wl0[16 + col0];     b.y = wl1[16 + col0];     acc1 = wmma4(a1, b, acc1);
        b.x = wl0[c2off] * msk2;  b.y = wl1[c2off] * msk2;  acc2 = wmma4(a1, b, acc2);
        b.x = wr0[col0];          b.y = wr1[col0];          acc0 = wmma4(a2, b, acc0);
        b.x = wr0[16 + col0];     b.y = wr1[16 + col0];     acc1 = wmma4(a2, b, acc1);
        b.x = wr0[c2off] * msk2;  b.y = wr1[c2off] * msk2;  acc2 = wmma4(a2, b, acc2);
    }
    int rowbase = rb * 16 + (lane >> 4) * 8;
    v8f accs[3] = {acc0, acc1, acc2};
    #pragma unroll
    for (int ct = 0; ct < 3; ++ct) {
        int col = ct * 16 + col0;
        if (col < 40) {
            float bn = bias[col];
            #pragma unroll
            for (int j = 0; j < 8; ++j) {
                int row = rowbase + j;
                if (row < nnodes) Logits[(long)row * 40 + col] = accs[ct][j] + bn;
            }
        }
    }
}

// ---------------- softmax over 40 cols, one wave per row ----------------
__global__ void softmax_kernel(const float* __restrict__ Logits, float* __restrict__ Out, int nnodes) {
    int wave = (blockIdx.x * blockDim.x + threadIdx.x) >> 5;
    int lane = threadIdx.x & 31;
    if (wave >= nnodes) return;
    const float* row = Logits + (long)wave * 40;
    float v0 = row[lane];                       // lanes 0..31 all valid (<40)
    bool hi = (lane + 32) < 40;                 // lanes 0..7
    float v1 = hi ? row[lane + 32] : -3.4e38f;
    float mx = fmaxf(v0, v1);
    for (int o = 16; o > 0; o >>= 1) mx = fmaxf(mx, __shfl_xor(mx, o, 32));
    float e0 = expf(v0 - mx);
    float e1 = hi ? expf(v1 - mx) : 0.0f;
    float s = e0 + e1;
    for (int o = 16; o > 0; o >>= 1) s += __shfl_xor(s, o, 32);
    float inv = 1.0f / s;
    Out[(long)wave * 40 + lane] = e0 * inv;
    if (hi) Out[(long)wave * 40 + lane + 32] = e1 * inv;
}

extern "C" void kernel_launch(void* const* d_in, const int* in_sizes, int n_in,
                              void* d_out, int out_size, void* d_ws, size_t ws_size,
                              hipStream_t stream) {
    const float* x    = (const float*)d_in[0];
    const float* Wl1  = (const float*)d_in[1];
    const float* Wr1  = (const float*)d_in[2];
    const float* b1   = (const float*)d_in[3];
    const float* Wl2  = (const float*)d_in[4];
    const float* Wr2  = (const float*)d_in[5];
    const float* b2   = (const float*)d_in[6];
    const int* esrc   = (const int*)d_in[7];
    const int* edst   = (const int*)d_in[8];

    const int H  = in_sizes[3];           // 64
    const int D  = in_sizes[1] / H;       // 64
    const int Nn = in_sizes[0] / D;       // 100000
    const int E  = in_sizes[7];           // 1600000
    (void)n_in; (void)out_size; (void)ws_size;

    float* ws     = (float*)d_ws;
    float* deg    = ws;                        // N
    float* dis    = ws + (long)Nn;             // N
    float* cntinv = ws + 2L * Nn;              // N
    float* bufA   = ws + 3L * Nn;              // N*64
    float* bufB   = bufA + (long)Nn * 64;      // N*64
    float* bufC   = bufB + (long)Nn * 64;      // N*64

    const int TB = 256;
    const size_t featBytes = (size_t)Nn * 64 * sizeof(float);
    dim3 gE((E + TB - 1) / TB), gN((Nn + TB - 1) / TB);
    dim3 gProp(((long)E * 16 + TB - 1) / TB);
    int rbCount = (Nn + 15) / 16;
    dim3 gGemm(((long)rbCount * 32 + TB - 1) / TB);
    dim3 gSm(((long)Nn * 32 + TB - 1) / TB);

    // 1) degree
    hipMemsetAsync(deg, 0, (size_t)Nn * sizeof(float), stream);
    deg_kernel<<<gE, TB, 0, stream>>>(edst, deg, E);
    norm_kernel<<<gN, TB, 0, stream>>>(deg, dis, cntinv, Nn);

    // 2) KProp step 1: x -> bufA (weighted)
    hipMemsetAsync(bufA, 0, featBytes, stream);
    prop_kernel<<<gProp, TB, 0, stream>>>(x, bufA, esrc, edst, dis, E, 1);
    // 3) KProp step 2: bufA -> bufB (weighted). h = bufB
    hipMemsetAsync(bufB, 0, featBytes, stream);
    prop_kernel<<<gProp, TB, 0, stream>>>(bufA, bufB, esrc, edst, dis, E, 1);
    // 4) agg1 raw sums: h -> bufA (plain)
    hipMemsetAsync(bufA, 0, featBytes, stream);
    prop_kernel<<<gProp, TB, 0, stream>>>(bufB, bufA, esrc, edst, dis, E, 0);
    // 5) GEMM1 + SELU -> h1 = bufC
    gemm1_kernel<<<gGemm, TB, 0, stream>>>(bufA, bufB, cntinv, Wl1, Wr1, b1, bufC, Nn);
    // 6) agg2 raw sums: h1 -> bufA (plain)
    hipMemsetAsync(bufA, 0, featBytes, stream);
    prop_kernel<<<gProp, TB, 0, stream>>>(bufC, bufA, esrc, edst, dis, E, 0);
    // 7) GEMM2 -> logits in bufB (reused, stride 40)
    gemm2_kernel<<<gGemm, TB, 0, stream>>>(bufA, bufC, cntinv, Wl2, Wr2, b2, bufB, Nn);
    // 8) softmax -> d_out
    softmax_kernel<<<gSm, TB, 0, stream>>>(bufB, (float*)d_out, Nn);
}